// QuaternionBatchNorm_60584808677741
// MI455X (gfx1250) — compile-verified
//
#include <hip/hip_runtime.h>

// Quaternion BatchNorm for x:[32, 4*64, 64, 64] fp32, 3-pass, deterministic.
// P1: per-(b,c) moment partials (16 samples/thread, shuffle+LDS reduce).
// P2: one wave per channel: butterfly-total stats, Cholesky chain, M = G*W
//     via V_WMMA_F32_16X16X4_F32 (K=4 exactly matches the 4x4x4 product).
// P3: out = M*x + (beta - M*mean); float4 streams, NT stores preserve x in L2.

#define QBN_EPS 1e-5f
#define C_ 64
#define B_ 32
#define HW_ 4096
#define COMP_STRIDE (C_ * HW_)        // 262144 floats between quaternion planes
#define BATCH_STRIDE (4 * C_ * HW_)   // 1048576 floats per batch image
#define SLICES 32                     // one partial per (b, c) plane
#define NSAMP 131072.0f               // B_*HW_ samples per channel

typedef float v2f __attribute__((ext_vector_type(2)));
typedef float v4f __attribute__((ext_vector_type(4)));
typedef float v8f __attribute__((ext_vector_type(8)));

#define QACC(RR, II, JJ, KK)                                              \
  do {                                                                    \
    s0 += (RR); s1 += (II); s2 += (JJ); s3 += (KK);                       \
    s4 = fmaf((RR), (RR), s4); s5 = fmaf((II), (II), s5);                 \
    s6 = fmaf((JJ), (JJ), s6); s7 = fmaf((KK), (KK), s7);                 \
    s8 = fmaf((RR), (II), s8); s9 = fmaf((RR), (JJ), s9);                 \
    s10 = fmaf((RR), (KK), s10); s11 = fmaf((II), (JJ), s11);             \
    s12 = fmaf((II), (KK), s12); s13 = fmaf((JJ), (KK), s13);             \
  } while (0)

// ---------------------------------------------------------------- kernel 1
// grid = (B_, C_), block = 256. Each block covers one (b, c) HW plane of 4096
// samples; each thread accumulates 16 samples (4 coalesced float4 groups).
__global__ __launch_bounds__(256) void qbn_stats(const float* __restrict__ x,
                                                 float* __restrict__ ws1) {
  const int b   = blockIdx.x;   // 0..31
  const int c   = blockIdx.y;   // 0..63
  const int tid = threadIdx.x;  // 0..255

  float s0 = 0.f, s1 = 0.f, s2 = 0.f, s3 = 0.f, s4 = 0.f, s5 = 0.f, s6 = 0.f,
        s7 = 0.f, s8 = 0.f, s9 = 0.f, s10 = 0.f, s11 = 0.f, s12 = 0.f,
        s13 = 0.f;

  const int plane = b * BATCH_STRIDE + c * HW_;
#pragma unroll
  for (int u = 0; u < 4; ++u) {
    const int base = plane + (u << 10) + (tid << 2);
    const v4f r = *(const v4f*)(x + base);
    const v4f i = *(const v4f*)(x + base + COMP_STRIDE);
    const v4f j = *(const v4f*)(x + base + 2 * COMP_STRIDE);
    const v4f k = *(const v4f*)(x + base + 3 * COMP_STRIDE);
    QACC(r[0], i[0], j[0], k[0]);
    QACC(r[1], i[1], j[1], k[1]);
    QACC(r[2], i[2], j[2], k[2]);
    QACC(r[3], i[3], j[3], k[3]);
  }

  float acc[14] = {s0, s1, s2, s3, s4, s5, s6, s7, s8, s9, s10, s11, s12, s13};

  // wave32 butterfly (deterministic)
#pragma unroll
  for (int m = 16; m >= 1; m >>= 1) {
#pragma unroll
    for (int s = 0; s < 14; ++s) acc[s] += __shfl_xor(acc[s], m, 32);
  }

  __shared__ float sp[8][14];
  const int wave = tid >> 5;
  const int lane = tid & 31;
  if (lane == 0) {
#pragma unroll
    for (int s = 0; s < 14; ++s) sp[wave][s] = acc[s];
  }
  __syncthreads();
  if (tid < 14) {
    float v = 0.f;
#pragma unroll
    for (int w = 0; w < 8; ++w) v += sp[w][tid];
    ws1[(tid * C_ + c) * SLICES + b] = v;  // stat-major, coalesced in pass 2
  }
}

// ---------------------------------------------------------------- kernel 2
// grid = C_, block = 32 (one wave, EXEC all ones -> WMMA legal). Butterfly
// totals leave full sums in every lane; Cholesky chain; M = G*W via WMMA.
__device__ __forceinline__ float sel4(int r, float a, float b, float c,
                                      float d) {
  return (r == 0) ? a : ((r == 1) ? b : ((r == 2) ? c : d));
}

__global__ __launch_bounds__(32) void qbn_coef(
    const float* __restrict__ ws1, const float* __restrict__ g_rr,
    const float* __restrict__ g_ii, const float* __restrict__ g_jj,
    const float* __restrict__ g_kk, const float* __restrict__ g_ri,
    const float* __restrict__ g_rj, const float* __restrict__ g_rk,
    const float* __restrict__ g_ij, const float* __restrict__ g_ik,
    const float* __restrict__ g_jk, const float* __restrict__ beta,
    float* __restrict__ coef) {
  const int c   = blockIdx.x;
  const int tid = threadIdx.x;  // 0..31

  float S[14];
#pragma unroll
  for (int s = 0; s < 14; ++s) S[s] = ws1[(s * C_ + c) * SLICES + tid];
  // butterfly: every lane ends with the full per-channel sums
#pragma unroll
  for (int m = 16; m >= 1; m >>= 1) {
#pragma unroll
    for (int s = 0; s < 14; ++s) S[s] += __shfl_xor(S[s], m, 32);
  }

  const float inv_n = 1.0f / NSAMP;
  const float mr = S[0] * inv_n, mi = S[1] * inv_n, mj = S[2] * inv_n,
              mk = S[3] * inv_n;
  const float var_r = S[4] * inv_n - mr * mr + QBN_EPS;
  const float var_i = S[5] * inv_n - mi * mi + QBN_EPS;
  const float var_j = S[6] * inv_n - mj * mj + QBN_EPS;
  const float var_k = S[7] * inv_n - mk * mk + QBN_EPS;
  const float cov_ri = S[8] * inv_n - mr * mi;
  const float cov_rj = S[9] * inv_n - mr * mj;
  const float cov_rk = S[10] * inv_n - mr * mk;
  const float cov_ij = S[11] * inv_n - mi * mj;
  const float cov_ik = S[12] * inv_n - mi * mk;
  const float cov_jk = S[13] * inv_n - mj * mk;

  // Cholesky-style chain, mirroring the reference exactly
  const float w_rr = sqrtf(var_r);
  const float w_ri = cov_ri / w_rr;
  const float w_ii = sqrtf(var_i - w_ri * w_ri);
  const float w_rj = cov_rj / w_rr;
  const float w_ij = (cov_ij - w_ri * w_rj) / w_ii;
  const float w_jj = sqrtf(var_j - (w_ij * w_ij + w_rj * w_rj));
  const float w_rk = cov_rk / w_rr;
  const float w_ik = (cov_ik - w_ri * w_rk) / w_ii;
  const float w_jk = (cov_jk - (w_ij * w_ik + w_rj * w_rk)) / w_jj;
  const float w_kk = sqrtf(var_k - (w_jk * w_jk + w_ik * w_ik + w_rk * w_rk));

  const float grr = g_rr[c], gii = g_ii[c], gjj = g_jj[c], gkk = g_kk[c];
  const float gri = g_ri[c], grj = g_rj[c], grk = g_rk[c];
  const float gij = g_ij[c], gik = g_ik[c], gjk = g_jk[c];

  const int row = tid & 15;
  const int hi  = tid >> 4;  // A vgprs carry K=2*hi, 2*hi+1; B vgprs rows hi, 2+hi
  const bool live = (row < 4);

  // A = G (symmetric), 16x4 layout: lane(row,hi): a0=G[row][2hi], a1=G[row][2hi+1]
  v2f Av;
  Av[0] = live ? (hi ? sel4(row, grj, gij, gjj, gjk)   // G col 2
                     : sel4(row, grr, gri, grj, grk))  // G col 0
               : 0.f;
  Av[1] = live ? (hi ? sel4(row, grk, gik, gjk, gkk)   // G col 3
                     : sel4(row, gri, gii, gij, gik))  // G col 1
               : 0.f;
  // B = W (upper-tri), 4x16 layout: b0 = rows 0/1 striped, b1 = rows 2/3
  v2f Bv;
  Bv[0] = live ? (hi ? sel4(row, 0.f, w_ii, w_ij, w_ik)     // W row 1
                     : sel4(row, w_rr, w_ri, w_rj, w_rk))   // W row 0
               : 0.f;
  Bv[1] = live ? (hi ? sel4(row, 0.f, 0.f, 0.f, w_kk)       // W row 3
                     : sel4(row, 0.f, 0.f, w_jj, w_jk))     // W row 2
               : 0.f;

#if __has_builtin(__builtin_amdgcn_wmma_f32_16x16x4_f32)
  v8f Cz = {};
  v8f D = __builtin_amdgcn_wmma_f32_16x16x4_f32(false, Av, false, Bv, (short)0,
                                                Cz, false, false);
  float M0 = D[0], M1 = D[1], M2 = D[2], M3 = D[3];  // lane n<4: M[j][n]
#else
  float M0 = 0.f, M1 = 0.f, M2 = 0.f, M3 = 0.f;
  if (tid < 4) {
    const float Gm[4][4] = {{grr, gri, grj, grk},
                            {gri, gii, gij, gik},
                            {grj, gij, gjj, gjk},
                            {grk, gik, gjk, gkk}};
    const float Wm[4][4] = {{w_rr, w_ri, w_rj, w_rk},
                            {0.f, w_ii, w_ij, w_ik},
                            {0.f, 0.f, w_jj, w_jk},
                            {0.f, 0.f, 0.f, w_kk}};
    float acc[4] = {0.f, 0.f, 0.f, 0.f};
#pragma unroll
    for (int p = 0; p < 4; ++p)
#pragma unroll
      for (int q = 0; q < 4; ++q) acc[p] += Gm[p][q] * Wm[q][tid];
    M0 = acc[0]; M1 = acc[1]; M2 = acc[2]; M3 = acc[3];
  }
#endif

  // lane n<4 holds column n of M; write M columns
  if (tid < 4) {
    coef[c * 20 + 0 * 4 + tid] = M0;
    coef[c * 20 + 1 * 4 + tid] = M1;
    coef[c * 20 + 2 * 4 + tid] = M2;
    coef[c * 20 + 3 * 4 + tid] = M3;
  }

  // bias_p = beta_p - sum_q M[p][q]*m_q : per-lane column term, xor-reduce 0..3
  const float m_n = sel4(row, mr, mi, mj, mk);
  float t0 = M0 * m_n, t1 = M1 * m_n, t2 = M2 * m_n, t3 = M3 * m_n;
#pragma unroll
  for (int m = 1; m <= 2; m <<= 1) {
    t0 += __shfl_xor(t0, m, 32);
    t1 += __shfl_xor(t1, m, 32);
    t2 += __shfl_xor(t2, m, 32);
    t3 += __shfl_xor(t3, m, 32);
  }
  if (tid == 0) {
    coef[c * 20 + 16 + 0] = beta[0 * C_ + c] - t0;
    coef[c * 20 + 16 + 1] = beta[1 * C_ + c] - t1;
    coef[c * 20 + 16 + 2] = beta[2 * C_ + c] - t2;
    coef[c * 20 + 16 + 3] = beta[3 * C_ + c] - t3;
  }
}

// ---------------------------------------------------------------- kernel 3
// grid = B_*C_ = 2048, block = 256. Each block applies M,bias to one (b, c)
// plane (4096 samples); 4 float4 groups/thread amortize the 20 coef loads.
__global__ __launch_bounds__(256) void qbn_apply(const float* __restrict__ x,
                                                 const float* __restrict__ coef,
                                                 float* __restrict__ out) {
  const int bid = blockIdx.x;
  const int tid = threadIdx.x;
  const int c   = bid & 63;
  const int b   = bid >> 6;

  const float* cf = coef + c * 20;
  float M[16], bb[4];
#pragma unroll
  for (int t = 0; t < 16; ++t) M[t] = cf[t];
#pragma unroll
  for (int t = 0; t < 4; ++t) bb[t] = cf[16 + t];

  const int plane = b * BATCH_STRIDE + c * HW_;
#pragma unroll
  for (int u = 0; u < 4; ++u) {
    const int base = plane + (u << 10) + (tid << 2);
    const v4f r = *(const v4f*)(x + base);
    const v4f i = *(const v4f*)(x + base + COMP_STRIDE);
    const v4f j = *(const v4f*)(x + base + 2 * COMP_STRIDE);
    const v4f k = *(const v4f*)(x + base + 3 * COMP_STRIDE);

    v4f o0 = M[0] * r + M[1] * i + M[2] * j + M[3] * k + bb[0];
    v4f o1 = M[4] * r + M[5] * i + M[6] * j + M[7] * k + bb[1];
    v4f o2 = M[8] * r + M[9] * i + M[10] * j + M[11] * k + bb[2];
    v4f o3 = M[12] * r + M[13] * i + M[14] * j + M[15] * k + bb[3];

    __builtin_nontemporal_store(o0, (v4f*)(out + base));
    __builtin_nontemporal_store(o1, (v4f*)(out + base + COMP_STRIDE));
    __builtin_nontemporal_store(o2, (v4f*)(out + base + 2 * COMP_STRIDE));
    __builtin_nontemporal_store(o3, (v4f*)(out + base + 3 * COMP_STRIDE));
  }
}

// ---------------------------------------------------------------- launcher
extern "C" void kernel_launch(void* const* d_in, const int* in_sizes, int n_in,
                              void* d_out, int out_size, void* d_ws,
                              size_t ws_size, hipStream_t stream) {
  (void)in_sizes; (void)n_in; (void)out_size; (void)ws_size;
  const float* x    = (const float*)d_in[0];
  const float* grr  = (const float*)d_in[1];
  const float* gii  = (const float*)d_in[2];
  const float* gjj  = (const float*)d_in[3];
  const float* gkk  = (const float*)d_in[4];
  const float* gri  = (const float*)d_in[5];
  const float* grj  = (const float*)d_in[6];
  const float* grk  = (const float*)d_in[7];
  const float* gij  = (const float*)d_in[8];
  const float* gik  = (const float*)d_in[9];
  const float* gjk  = (const float*)d_in[10];
  const float* beta = (const float*)d_in[11];
  float* out = (float*)d_out;

  float* ws1  = (float*)d_ws;             // 14*64*32 floats (partials)
  float* coef = ws1 + 14 * C_ * SLICES;   // 64*20 floats (M + bias)

  dim3 g1(B_, C_);
  qbn_stats<<<g1, 256, 0, stream>>>(x, ws1);
  qbn_coef<<<C_, 32, 0, stream>>>(ws1, grr, gii, gjj, gkk, gri, grj, grk,
                                  gij, gik, gjk, beta, coef);
  qbn_apply<<<B_ * C_, 256, 0, stream>>>(x, coef, out);
}